// CustomSPPLayer_22694607192324
// MI455X (gfx1250) — compile-verified
//
#include <hip/hip_runtime.h>
#include <cstdint>

// SPP max-pool pyramid, MI455X (gfx1250, wave32).
// Input : x (16, 256, 64, 64) f32
// Output: (16, 5376, 13, 13) f32   [channels: 256 (lvl0) + 1024 (lvl1) + 4096 (lvl2)]
//
// One workgroup per (b,c) plane. Plane (16 KB) staged into LDS with
// GLOBAL_LOAD_ASYNC_TO_LDS_B128 (ASYNCcnt), then a 3-level max pyramid:
//   M4 : 16x16 grid of 4x4 block maxes       (each input element read once)
//   M8 : 15x15, max of 2x2 neighboring M4
//   out0: 13x13, max of M8[i|i+2][j|j+2]
// lvl1 cell(p,q) = M8[i+2p][j+2q], lvl2 cell(p,q) = M4[i+p][j+q].

#define H 64
#define W 64
#define NC 256
#define NB 16
#define OUT_HW 169          // 13*13
#define CH_TOTAL 5376       // 256 + 1024 + 4096

__launch_bounds__(256)
__global__ void spp_pyramid_kernel(const float* __restrict__ x,
                                   float* __restrict__ out) {
    const int plane = blockIdx.x;          // b*256 + c
    const int b = plane >> 8;
    const int c = plane & 255;
    const int t = threadIdx.x;             // 0..255 (8 waves)

    __shared__ float tile[H * W];          // 16 KB raw plane
    __shared__ float m4[16 * 16];          // 4x4 block maxes
    __shared__ float m8[15 * 16];          // 8x8 maxes (stride-16 padded)

    // ---- Stage the 64x64 plane into LDS via async global->LDS DMA ----
    {
        const uint32_t lds0  = (uint32_t)(uintptr_t)&tile[0];      // LDS byte offset
        const uint64_t gbase = (uint64_t)(uintptr_t)(x + (size_t)plane * (H * W));
#pragma unroll
        for (int r = 0; r < 4; ++r) {
            const uint32_t off  = (uint32_t)(r * 256 + t) * 16u;   // bytes (b128/lane)
            const uint32_t lAdr = lds0 + off;
            const uint64_t gAdr = gbase + off;
            asm volatile("global_load_async_to_lds_b128 %0, %1, off"
                         :: "v"(lAdr), "v"(gAdr) : "memory");
        }
        asm volatile("s_wait_asynccnt 0x0" ::: "memory");
    }
    __syncthreads();

    // ---- M4: each thread reduces one 4x4 block (4 x ds_load_b128) ----
    {
        const int m = t >> 4, n = t & 15;
        const float4* trow = (const float4*)tile;   // 16 float4 per 64-float row
        float mx;
        {
            float4 v = trow[(4 * m) * 16 + n];
            mx = fmaxf(fmaxf(v.x, v.y), fmaxf(v.z, v.w));
        }
#pragma unroll
        for (int r = 1; r < 4; ++r) {
            float4 v = trow[(4 * m + r) * 16 + n];
            mx = fmaxf(mx, fmaxf(fmaxf(v.x, v.y), fmaxf(v.z, v.w)));
        }
        m4[m * 16 + n] = mx;
    }
    __syncthreads();

    // ---- M8: 15x15, max of 2x2 M4 ----
    if (t < 225) {
        const int mm = t / 15, nn = t % 15;
        const float a = fmaxf(m4[mm * 16 + nn],       m4[mm * 16 + nn + 1]);
        const float d = fmaxf(m4[(mm + 1) * 16 + nn], m4[(mm + 1) * 16 + nn + 1]);
        m8[mm * 16 + nn] = fmaxf(a, d);
    }
    __syncthreads();

    // ---- Emit all three levels; 169 lanes, contiguous 169-float runs ----
    if (t < OUT_HW) {
        const int i = t / 13, j = t % 13;
        float* ob = out + (size_t)b * ((size_t)CH_TOTAL * OUT_HW);

        // level 0: 16x16 window max = max of 2x2 M8 at stride 2
        const float l0 = fmaxf(fmaxf(m8[i * 16 + j],       m8[i * 16 + j + 2]),
                               fmaxf(m8[(i + 2) * 16 + j], m8[(i + 2) * 16 + j + 2]));
        ob[(size_t)c * OUT_HW + t] = l0;

        // level 1: ch = 256 + c*4 + p*2 + q, value = M8[i+2p][j+2q]
#pragma unroll
        for (int p = 0; p < 2; ++p)
#pragma unroll
            for (int q = 0; q < 2; ++q) {
                const size_t ch = 256 + (size_t)c * 4 + p * 2 + q;
                ob[ch * OUT_HW + t] = m8[(i + 2 * p) * 16 + (j + 2 * q)];
            }

        // level 2: ch = 1280 + c*16 + p*4 + q, value = M4[i+p][j+q]
#pragma unroll
        for (int p = 0; p < 4; ++p)
#pragma unroll
            for (int q = 0; q < 4; ++q) {
                const size_t ch = 1280 + (size_t)c * 16 + p * 4 + q;
                ob[ch * OUT_HW + t] = m4[(i + p) * 16 + (j + q)];
            }
    }
}

extern "C" void kernel_launch(void* const* d_in, const int* in_sizes, int n_in,
                              void* d_out, int out_size, void* d_ws, size_t ws_size,
                              hipStream_t stream) {
    (void)in_sizes; (void)n_in; (void)d_ws; (void)ws_size; (void)out_size;
    const float* x = (const float*)d_in[0];
    float* out = (float*)d_out;
    spp_pyramid_kernel<<<NB * NC, 256, 0, stream>>>(x, out);
}